// ShiftedConv2d_25872882991120
// MI455X (gfx1250) — compile-verified
//
#include <hip/hip_runtime.h>

// ShiftedConv2d reduces (see analysis) to:
//   out[:] = 0
//   out[(n*16+s)*16384] = tens[n,127,127] * filters[s,0,0,0]  iff shifts[s]==(7,7)
// for n in [0,256), s in [0,16).  Dominant cost: 256MB of zero stores.

typedef __attribute__((ext_vector_type(2))) float v2f;
typedef __attribute__((ext_vector_type(4))) float v4f;
typedef __attribute__((ext_vector_type(8))) float v8f;

// ---------------------------------------------------------------------------
// Kernel 1: non-temporal wide zero fill (global_store_b128, NT hint).
// Output (256MB) > L2 (192MB), so bypassing the cache hierarchy is correct.
// ---------------------------------------------------------------------------
__global__ void zero_fill_b128(v4f* __restrict__ out, long long n4) {
    long long i      = (long long)blockIdx.x * blockDim.x + threadIdx.x;
    long long stride = (long long)gridDim.x * blockDim.x;
    v4f z = {0.f, 0.f, 0.f, 0.f};
    for (; i < n4; i += stride) {
        __builtin_nontemporal_store(z, out + i);
    }
}

// ---------------------------------------------------------------------------
// Kernel 2: corner outer product via V_WMMA_F32_16X16X4_F32.
// One wave (32 lanes, EXEC all-ones). 16 WMMAs cover n = 0..255 in M-tiles of
// 16; columns are the S=16 shift channels.
//   A (16x4): A[m,0] = tens[tile*16+m, 127,127]; other K columns zero.
//   B (4x16): B[0,s] = (shifts[s]==(7,7)) ? filters[s,0,0,0] : 0; rest zero.
//   D (16x16): D[m,s] -> out[((tile*16+m)*16 + s) * 16384]
// A layout: lane = 16*(k>=2)+m, vgpr = k&1. D layout: lanes 0-15 M=v,
// lanes 16-31 M=8+v, N = lane&15.
// ---------------------------------------------------------------------------
__global__ void corner_wmma(const float* __restrict__ tens,
                            const float* __restrict__ filters,
                            const int*   __restrict__ shifts,
                            float* __restrict__ out) {
    const int  lane = threadIdx.x;     // 0..31, exactly one wave32
    const int  l15  = lane & 15;
    const bool lo   = lane < 16;

    // Build B: only K=0 row (lanes 0-15, VGPR 0) is nonzero.
    const float fv   = filters[l15 * 49];                 // filters[s,0,0,0]
    const int   shh  = shifts[2 * l15 + 0];
    const int   shw  = shifts[2 * l15 + 1];
    const float gate = (shh == 7 && shw == 7) ? 1.0f : 0.0f;
    v2f b;
    b.x = lo ? fv * gate : 0.0f;   // K=0 (lanes 0-15) / K=2 (lanes 16-31, zero)
    b.y = 0.0f;                    // K=1 / K=3

    for (int tile = 0; tile < 16; ++tile) {
        const int   n = tile * 16 + l15;                  // row (M) index
        const float t = tens[(size_t)n * 16384 + 16383];  // tens[n,127,127]
        v2f a;
        a.x = lo ? t : 0.0f;   // K=0 column holds t; all other K zero
        a.y = 0.0f;
        v8f c = {};
        // (neg_a, A, neg_b, B, c_mod, C, reuse_a, reuse_b)
        c = __builtin_amdgcn_wmma_f32_16x16x4_f32(
                false, a, false, b, (short)0, c, false, false);

        // Scatter D: lane half selects M base; column = l15 (the s index).
        const int sIdx  = l15;
        const int mBase = lo ? 0 : 8;
#pragma unroll
        for (int v = 0; v < 8; ++v) {
            const int    m   = mBase + v;
            const size_t off = ((size_t)(tile * 16 + m) * 16 + sIdx) * 16384;
            out[off] = c[v];
        }
    }
}

extern "C" void kernel_launch(void* const* d_in, const int* in_sizes, int n_in,
                              void* d_out, int out_size, void* d_ws, size_t ws_size,
                              hipStream_t stream) {
    const float* tens    = (const float*)d_in[0];  // (8,32,128,128) f32
    const float* filters = (const float*)d_in[1];  // (16,1,7,7)     f32
    const int*   shifts  = (const int*)  d_in[2];  // (16,2)         i32
    float*       out     = (float*)d_out;          // (8,512,128,128) f32

    // 67,108,864 floats -> 16,777,216 float4 stores.
    const long long n4 = (long long)out_size / 4;
    const int threads = 256;
    const int blocks  = 8192;   // grid-stride; ~8 B128 NT stores per thread
    zero_fill_b128<<<blocks, threads, 0, stream>>>((v4f*)out, n4);

    // One wave writes the 4096 (mostly zero) corner products via WMMA.
    corner_wmma<<<1, 32, 0, stream>>>(tens, filters, shifts, out);
}